// BodyFst_67018669687230
// MI455X (gfx1250) — compile-verified
//
#include <hip/hip_runtime.h>
#include <hip/hip_bf16.h>

// CDNA5 (gfx1250) stereo cost-volume via V_WMMA_F32_16X16X4_F32.
//
// cost[j,h,x] = sum_c L[c,h,x] * R[c,h,x-j]   (0 when x<j),  C = 32.
//
// Tile mapping: A[m,k] = L[k, x0+m], B[k,n] = R[k, x0-16t+n]  =>
// D[m,n] = cost[j = 16t+m-n, x = x0+m]. Eight chained K=4 WMMAs cover C=32.
// Each valid (j,x) comes from exactly one (t,n), so tiles are disjoint.
// Results are staged in LDS (de-diagonalized) and streamed out as 128B-
// contiguous 16B-vector non-temporal stores: the kernel is store-BW bound
// (~115MB out vs 1.84 GFLOP => HBM roofline ~6.5us at 23.3 TB/s).
//
// Branch-free inner loop:
//  - LDS has 16 pad rows on each side of the D valid rows; out-of-band tile
//    elements (j<0 or j>=D) are stored unconditionally into pad rows that are
//    never read back -> no per-element compare/exec-mask juggling.
//  - Right-image columns are clamped to 0 and the value selected to 0.0 with
//    a ternary -> v_max + v_cndmask instead of exec save/restore branches.

typedef float v2f __attribute__((ext_vector_type(2)));
typedef float v4f __attribute__((ext_vector_type(4)));
typedef float v8f __attribute__((ext_vector_type(8)));

#define MAX_D  192
#define PAD_J  16   // pad rows above and below the valid j range
#define ROW_W  36   // LDS row stride (words): 32 data + 4 pad; 16B-aligned, bank-friendly

__global__ __launch_bounds__(128) void cost_volume_wmma(
    const float* __restrict__ L, const float* __restrict__ R,
    float* __restrict__ out, int H, int W, int D)
{
    __shared__ __align__(16) float cost[(MAX_D + 2 * PAD_J) * ROW_W];  // 31.5 KB

    const int tid   = threadIdx.x;
    const int lane  = tid & 31;
    const int wave  = tid >> 5;       // 0..3
    const int bh    = wave >> 1;      // which 16-wide x half of the 32-wide block
    const int wsub  = wave & 1;       // wave index within the half
    const int h     = blockIdx.y;
    const int x0    = blockIdx.x * 32 + bh * 16;   // this wave's left-image base x
    const long HW   = (long)H * W;
    const long rowB = (long)h * W;

    const int n16   = lane & 15;      // matrix row/col index held by this lane
    const int khalf = lane >> 4;      // 0: K={0,1}; 1: K={2,3} (f32 A/B VGPR layout)

    // ---- Preload A fragments once per wave: A[m,k] = L[c, h, x0+m].
    // Lane holds m = n16; VGPR pair holds channels c0, c0+1.
    v2f a[8];
    {
        const float* Lp = L + rowB + x0 + n16;
        #pragma unroll
        for (int ks = 0; ks < 8; ++ks) {
            const int c0 = 4 * ks + 2 * khalf;
            a[ks].x = Lp[(long)(c0    ) * HW];
            a[ks].y = Lp[(long)(c0 + 1) * HW];
        }
    }

    // ---- Diagonal tiles: tile t covers j in [16t-15, 16t+15].
    const int T = D / 16 + 1;
    for (int t = wsub; t < T; t += 2) {            // uniform per wave
        const int xc  = x0 - 16 * t + n16;         // right-image column for this lane
        const int xcc = xc < 0 ? 0 : xc;           // clamped (always in-bounds)
        v2f b[8];
        #pragma unroll
        for (int ks = 0; ks < 8; ++ks) {
            const int c0 = 4 * ks + 2 * khalf;
            float b0 = R[(long)(c0    ) * HW + rowB + xcc];
            float b1 = R[(long)(c0 + 1) * HW + rowB + xcc];
            b[ks].x = (xc >= 0) ? b0 : 0.0f;       // banded zero-pad (x < j)
            b[ks].y = (xc >= 0) ? b1 : 0.0f;
        }

        v8f acc = {0.f, 0.f, 0.f, 0.f, 0.f, 0.f, 0.f, 0.f};
        #pragma unroll
        for (int ks = 0; ks < 8; ++ks) {
            acc = __builtin_amdgcn_wmma_f32_16x16x4_f32(
                false, a[ks], false, b[ks], (short)0, acc, false, false);
        }

        // Unconditional scatter tile -> LDS [PAD_J + j][xl]. C/D layout:
        // VGPR r, lane l holds (m = r + 8*(l>=16), n = l&15); j = 16t+m-n.
        // j in [-15, D+15] always lands inside the padded buffer.
        const int base = (PAD_J + 16 * t - n16) * ROW_W + bh * 16;
        #pragma unroll
        for (int r = 0; r < 8; ++r) {
            const int m = r + 8 * khalf;
            cost[base + m * (ROW_W + 1)] = acc[r];   // (j)*ROW_W + xl, j=16t+m-n16
        }
    }

    __syncthreads();

    // ---- Coalesced streaming write-out: each j-row is 32 floats = 128B
    // contiguous in global memory; use non-temporal stores (pure streaming,
    // never re-read) so L2 keeps the reused L/R inputs instead.
    const int xbase = blockIdx.x * 32;
    for (int idx = tid; idx < D * 8; idx += 128) {
        const int j = idx >> 3;
        const int c = idx & 7;                      // 16B chunk within the row
        const v4f v = *(const v4f*)&cost[(PAD_J + j) * ROW_W + c * 4];
        v4f* dst = (v4f*)(out + (long)j * HW + rowB + xbase + c * 4);
        __builtin_nontemporal_store(v, dst);
    }
}

extern "C" void kernel_launch(void* const* d_in, const int* in_sizes, int n_in,
                              void* d_out, int out_size, void* d_ws, size_t ws_size,
                              hipStream_t stream) {
    (void)in_sizes; (void)n_in; (void)d_ws; (void)ws_size; (void)out_size;

    const float* L0 = (const float*)d_in[0];
    const float* R0 = (const float*)d_in[1];
    const float* L1 = (const float*)d_in[2];
    const float* R1 = (const float*)d_in[3];
    const float* L2 = (const float*)d_in[4];
    const float* R2 = (const float*)d_in[5];

    float* out = (float*)d_out;

    // Scale 0: (1,32,256,512), D=192
    cost_volume_wmma<<<dim3(512 / 32, 256), 128, 0, stream>>>(L0, R0, out, 256, 512, 192);
    out += (long)192 * 256 * 512;

    // Scale 1: (1,32,128,256), D=96
    cost_volume_wmma<<<dim3(256 / 32, 128), 128, 0, stream>>>(L1, R1, out, 128, 256, 96);
    out += (long)96 * 128 * 256;

    // Scale 2: (1,32,64,128), D=48
    cost_volume_wmma<<<dim3(128 / 32, 64), 128, 0, stream>>>(L2, R2, out, 64, 128, 48);
}